// TreePositionalEncodings_30039001268275
// MI455X (gfx1250) — compile-verified
//
#include <hip/hip_runtime.h>
#include <cmath>

// Tree positional encodings as a single-pass f32 WMMA "GEMM":
//   out[m, j] = sum_k Ppad[m, k] * W[k, j]
// M = 32*2048 rows (zero-padded for sos/tail), K = 20, N = 1024.
// W is column-one-hot in K (W[k,j] != 0 iff j/51 == k).
// Prepass builds the 1024-entry per-column weight table (all transcendentals)
// so the hot kernel is pure {load, select, wmma, NT-store} and stays under the
// 268 MB / 23.3 TB/s ~= 11.5 us write-bandwidth floor.

typedef __attribute__((ext_vector_type(2))) float v2f;
typedef __attribute__((ext_vector_type(8))) float v8f;

#define BS      32
#define NNODES  1364      // 4 + 16 + 64 + 256 + 1024
#define KDIM    20        // depth * width
#define DMODEL  1024
#define MAXLEN  2048
#define NFEAT   51        // DMODEL / KDIM

// ---- prepass: wtab[col] = tanh(p[col%51])^(col/51/4) * sqrt((1-t^2)*512) ----
__global__ void build_wtab(const float* __restrict__ p, float* __restrict__ wtab)
{
    const int col = blockIdx.x * blockDim.x + threadIdx.x;
    if (col >= DMODEL) return;
    const int f  = col % NFEAT;
    const int ka = col / NFEAT;              // active K row; >=20 => pad column
    const float tp  = tanhf(p[f]);
    const float nrm = sqrtf((1.0f - tp * tp) * (0.5f * (float)DMODEL));
    const int lvl = ka >> 2;                 // tree level = k / width
    float w = nrm;
    for (int i = 0; i < lvl && i < 5; ++i) w *= tp;
    wtab[col] = (ka < KDIM) ? w : 0.0f;
}

__global__ __launch_bounds__(256) void tree_pos_enc_wmma(
    const float* __restrict__ positions,
    const float* __restrict__ wtab,
    float* __restrict__ out)
{
    const int lane = threadIdx.x & 31;
    const int waveInBlk = threadIdx.x >> 5;
    const int unit = blockIdx.x * 8 + waveInBlk;  // 65536 wave-units
    const int mt = unit >> 4;                     // 0..4095  : 16-row M tile
    const int ng = unit & 15;                     // 0..15    : 64-col N group

    const int half = lane >> 4;                   // A/B K-slot base selector
    const int lm   = lane & 15;

    // ---- A fragments: 16x4 f32 layout, lanes 0-15 hold K=0/1, 16-31 K=2/3 ----
    const int m = mt * 16 + lm;                   // flat row: b*2048 + s
    const int s = m & (MAXLEN - 1);
    const int b = m >> 11;
    const int n = s - 1;                          // node index (s==0 -> sos)
    const bool valid = (unsigned)n < (unsigned)NNODES;
    const float* prow =
        positions + ((size_t)b * NNODES + (size_t)(valid ? n : 0)) * KDIM;

    float a0[5], a1[5];
#pragma unroll
    for (int c = 0; c < 5; ++c) {
        const int k0 = 4 * c + 2 * half;
        a0[c] = valid ? prow[k0]     : 0.0f;
        a1[c] = valid ? prow[k0 + 1] : 0.0f;
    }

#pragma unroll
    for (int t = 0; t < 4; ++t) {
        const int colBase = ng * 64 + t * 16;
        const int col = colBase + lm;             // output feature column
        const float wval = wtab[col];             // 4 KB L2/L0-hot table
        const int ka = col / NFEAT;               // active K row (one-hot in K)

        v8f acc = {0.f, 0.f, 0.f, 0.f, 0.f, 0.f, 0.f, 0.f};
#pragma unroll
        for (int c = 0; c < 5; ++c) {
            const int k0 = 4 * c + 2 * half;
            v2f av; av[0] = a0[c]; av[1] = a1[c];
            v2f bv;
            bv[0] = (k0     == ka) ? wval : 0.0f;
            bv[1] = (k0 + 1 == ka) ? wval : 0.0f;
            // D = A(16x4,f32) x B(4x16,f32) + C ; exact in f32
            acc = __builtin_amdgcn_wmma_f32_16x16x4_f32(
                      false, av, false, bv, (short)0, acc, false, false);
        }

        // C layout: VGPR v -> lanes 0-15: (M=v, N=lane), lanes 16-31: (M=v+8)
#pragma unroll
        for (int v = 0; v < 8; ++v) {
            const int row = mt * 16 + v + 8 * half;
            __builtin_nontemporal_store(
                acc[v], out + (size_t)row * DMODEL + (size_t)(colBase + lm));
        }
    }
}

extern "C" void kernel_launch(void* const* d_in, const int* in_sizes, int n_in,
                              void* d_out, int out_size, void* d_ws, size_t ws_size,
                              hipStream_t stream) {
    (void)in_sizes; (void)n_in; (void)ws_size; (void)out_size;
    const float* p         = (const float*)d_in[0];   // [64] f32
    const float* positions = (const float*)d_in[1];   // [32,1364,20] f32
    float* out  = (float*)d_out;                      // [32,2048,1024] f32
    float* wtab = (float*)d_ws;                       // 1024 f32 = 4 KB scratch

    build_wtab<<<(DMODEL + 255) / 256, 256, 0, stream>>>(p, wtab);

    // 4096 M-tiles * 16 N-groups = 65536 waves ; 8 waves (256 thr) per block
    const int numBlocks = (BS * MAXLEN / 16) * 16 / 8;  // 8192
    tree_pos_enc_wmma<<<numBlocks, 256, 0, stream>>>(positions, wtab, out);
}